// Subunit_NN_91122026152750
// MI455X (gfx1250) — compile-verified
//
#include <hip/hip_runtime.h>
#include <hip/hip_bf16.h>

#define SUB_NO 5
#define T_NO   201
#define H_NO   64
#define T_DATA 50000
#define E_SYN  2000
#define I_SYN  200
#define NCH    (2*SUB_NO)      // 10 interleaved channels: even=E, odd=I

typedef float v2f __attribute__((ext_vector_type(2)));
typedef float v8f __attribute__((ext_vector_type(8)));

// ---- workspace layout (float offsets) ----
#define WS_BE   0                              // zero-padded B for E pass: [E_SYN][16]
#define WS_BI   (WS_BE + E_SYN*16)             // zero-padded B for I pass: [I_SYN][16]
#define WS_WEFF (WS_BI + I_SYN*16)             // folded conv weights: [SUB_NO][2][T_NO]
#define WS_BEFF (WS_WEFF + SUB_NO*2*T_NO)      // folded bias: [SUB_NO]
#define WS_SG   (WS_BEFF + 16)                 // grouped signal, channel-major: [NCH][T_DATA]

// ---------------------------------------------------------------------------
// Kernel 1: softmax over subunit axis + build zero-padded WMMA B matrices.
//   BE[k][n] = (n even, n<10) ? softmax(C_e)[n/2][k] : 0
//   BI[k][n] = (n odd , n<10) ? softmax(C_i)[n>>1][k] : 0
// ---------------------------------------------------------------------------
__global__ void build_b(const float* __restrict__ Ce_raw,
                        const float* __restrict__ Ci_raw,
                        const float* __restrict__ temp,
                        float* __restrict__ ws)
{
    const int idx = blockIdx.x * blockDim.x + threadIdx.x;
    const float invT = 1.0f / temp[0];

    if (idx < E_SYN) {
        float x[SUB_NO];
        float m = -3.4e38f;
        #pragma unroll
        for (int s = 0; s < SUB_NO; ++s) { x[s] = Ce_raw[s*E_SYN + idx] * invT; m = fmaxf(m, x[s]); }
        float sum = 0.0f;
        #pragma unroll
        for (int s = 0; s < SUB_NO; ++s) { x[s] = expf(x[s] - m); sum += x[s]; }
        const float inv = 1.0f / sum;
        float* row = ws + WS_BE + idx*16;
        #pragma unroll
        for (int n = 0; n < 16; ++n) {
            float v = 0.0f;
            if (n < NCH && (n & 1) == 0) v = x[n >> 1] * inv;
            row[n] = v;
        }
    } else if (idx < E_SYN + I_SYN) {
        const int k = idx - E_SYN;
        float x[SUB_NO];
        float m = -3.4e38f;
        #pragma unroll
        for (int s = 0; s < SUB_NO; ++s) { x[s] = Ci_raw[s*I_SYN + k] * invT; m = fmaxf(m, x[s]); }
        float sum = 0.0f;
        #pragma unroll
        for (int s = 0; s < SUB_NO; ++s) { x[s] = expf(x[s] - m); sum += x[s]; }
        const float inv = 1.0f / sum;
        float* row = ws + WS_BI + k*16;
        #pragma unroll
        for (int n = 0; n < 16; ++n) {
            float v = 0.0f;
            if (n < NCH && (n & 1) == 1) v = x[n >> 1] * inv;   // (n-1)/2 == n>>1 for odd n
            row[n] = v;
        }
    }
}

// ---------------------------------------------------------------------------
// Kernel 2: fold per-subunit Linear(H,1) into the conv filters.
//   W_eff[(s*2+c)*201+k] = sum_h ff_w[s,h] * conv_w[(s*H+h), c, k]
//   b_eff[s]             = ff_b[s] + sum_h ff_w[s,h] * conv_b[s*H+h]
// ---------------------------------------------------------------------------
__global__ void build_weff(const float* __restrict__ conv_w,
                           const float* __restrict__ conv_b,
                           const float* __restrict__ ff_w,
                           const float* __restrict__ ff_b,
                           float* __restrict__ ws)
{
    const int idx = blockIdx.x * blockDim.x + threadIdx.x;
    const int NW = SUB_NO * 2 * T_NO;   // 2010
    if (idx < NW) {
        const int s   = idx / (2*T_NO);
        const int rem = idx % (2*T_NO);  // c*201 + k
        float acc = 0.0f;
        for (int h = 0; h < H_NO; ++h)
            acc = fmaf(ff_w[s*H_NO + h], conv_w[(size_t)(s*H_NO + h)*(2*T_NO) + rem], acc);
        ws[WS_WEFF + idx] = acc;
    } else if (idx < NW + SUB_NO) {
        const int s = idx - NW;
        float acc = ff_b[s];
        for (int h = 0; h < H_NO; ++h)
            acc = fmaf(ff_w[s*H_NO + h], conv_b[s*H_NO + h], acc);
        ws[WS_BEFF + s] = acc;
    }
}

// ---------------------------------------------------------------------------
// Kernel 3: WMMA group-GEMM. One wave computes a 16(T) x 16(N) fp32 tile with
// V_WMMA_F32_16X16X4_F32; E and I passes accumulate into disjoint (even/odd)
// columns via the zero-padded B matrices. EXEC stays all-ones (no divergence
// until after the last WMMA). 3125 tiles exactly cover T=50000.
// ---------------------------------------------------------------------------
__global__ __launch_bounds__(160) void group_gemm_wmma(
    const float* __restrict__ Se, const float* __restrict__ Si,
    float* __restrict__ ws)
{
    const float* __restrict__ BE = ws + WS_BE;
    const float* __restrict__ BI = ws + WS_BI;
    float* __restrict__ Sg = ws + WS_SG;

    const int wave = blockIdx.x * 5 + (threadIdx.x >> 5);  // 625 blocks * 5 waves = 3125
    const int lane = threadIdx.x & 31;
    const int half = lane >> 4;    // 0: K k0..k0+1 / C rows 0..7 ; 1: K k0+2..k0+3 / C rows 8..15
    const int r    = lane & 15;    // A row (M) and B/C/D column (N)
    const int t0   = wave << 4;

    v8f acc = {};

    {   // excitatory pass: K = 2000
        const float* a_ptr = Se + (size_t)(t0 + r) * E_SYN + 2*half;
        const float* b_ptr = BE + (2*half)*16 + r;
        for (int k0 = 0; k0 < E_SYN; k0 += 4) {
            v2f a = *(const v2f*)(a_ptr + k0);      // {A[M][k0+2h], A[M][k0+2h+1]}
            v2f b;
            b.x = b_ptr[k0*16];                     // B[k0+2h  ][r]
            b.y = b_ptr[k0*16 + 16];                // B[k0+2h+1][r]
            acc = __builtin_amdgcn_wmma_f32_16x16x4_f32(
                false, a, false, b, (short)0, acc, false, false);
        }
    }
    {   // inhibitory pass: K = 200
        const float* a_ptr = Si + (size_t)(t0 + r) * I_SYN + 2*half;
        const float* b_ptr = BI + (2*half)*16 + r;
        for (int k0 = 0; k0 < I_SYN; k0 += 4) {
            v2f a = *(const v2f*)(a_ptr + k0);
            v2f b;
            b.x = b_ptr[k0*16];
            b.y = b_ptr[k0*16 + 16];
            acc = __builtin_amdgcn_wmma_f32_16x16x4_f32(
                false, a, false, b, (short)0, acc, false, false);
        }
    }

    // C/D layout: VGPR v, lanes 0-15 -> M=v, lanes 16-31 -> M=v+8; N = lane%16.
    if (r < NCH) {
        float* col = Sg + (size_t)r * T_DATA + t0 + 8*half;
        #pragma unroll
        for (int v = 0; v < 8; ++v) col[v] = acc[v];
    }
}

// ---------------------------------------------------------------------------
// Kernel 4: folded 201-tap conv over 10 channels + LeakyReLU heads.
// Signal window and folded weights staged in LDS (~26.3 KB of 320 KB/WGP).
// ---------------------------------------------------------------------------
#define TB 256
__global__ __launch_bounds__(TB) void conv_head(
    const float* __restrict__ ws,
    const float* __restrict__ root_w, const float* __restrict__ root_b,
    const float* __restrict__ V_o, float* __restrict__ out)
{
    __shared__ float wsh[SUB_NO*2*T_NO];
    __shared__ float bsh[SUB_NO];
    __shared__ float tile[NCH][TB + T_NO - 1];   // [10][456]

    const float* Weff = ws + WS_WEFF;
    const float* beff = ws + WS_BEFF;
    const float* Sg   = ws + WS_SG;

    for (int i = threadIdx.x; i < SUB_NO*2*T_NO; i += TB) wsh[i] = Weff[i];
    if (threadIdx.x < SUB_NO) bsh[threadIdx.x] = beff[threadIdx.x];

    const int tblk = blockIdx.x * TB;
    for (int j = 0; j < NCH; ++j) {
        for (int i = threadIdx.x; i < TB + T_NO - 1; i += TB) {
            const int t = tblk + i - (T_NO/2);
            tile[j][i] = (t >= 0 && t < T_DATA) ? Sg[(size_t)j*T_DATA + t] : 0.0f;
        }
    }
    __syncthreads();

    const int t = tblk + threadIdx.x;
    if (t >= T_DATA) return;

    float rootacc = 0.0f;
    #pragma unroll
    for (int s = 0; s < SUB_NO; ++s) {
        float acc = bsh[s];
        #pragma unroll
        for (int c = 0; c < 2; ++c) {
            const float* w = wsh + (s*2 + c)*T_NO;
            const float* x = &tile[2*s + c][threadIdx.x];
            for (int k = 0; k < T_NO; ++k)
                acc = fmaf(w[k], x[k], acc);
        }
        acc = (acc > 0.0f) ? acc : 0.01f * acc;          // LeakyReLU(0.01)
        rootacc = fmaf(acc, root_w[s], rootacc);
    }
    float rr = rootacc + root_b[0];
    rr = (rr > 0.0f) ? rr : 0.01f * rr;
    out[t] = rr + V_o[0];
}

// ---------------------------------------------------------------------------
extern "C" void kernel_launch(void* const* d_in, const int* in_sizes, int n_in,
                              void* d_out, int out_size, void* d_ws, size_t ws_size,
                              hipStream_t stream)
{
    const float* Se     = (const float*)d_in[0];
    const float* Si     = (const float*)d_in[1];
    const float* temp   = (const float*)d_in[2];
    const float* Ce_raw = (const float*)d_in[3];
    const float* Ci_raw = (const float*)d_in[4];
    const float* conv_w = (const float*)d_in[5];
    const float* conv_b = (const float*)d_in[6];
    const float* ff_w   = (const float*)d_in[7];
    const float* ff_b   = (const float*)d_in[8];
    const float* root_w = (const float*)d_in[9];
    const float* root_b = (const float*)d_in[10];
    const float* V_o    = (const float*)d_in[11];
    float* out = (float*)d_out;
    float* ws  = (float*)d_ws;

    build_b<<<(E_SYN + I_SYN + 255)/256, 256, 0, stream>>>(Ce_raw, Ci_raw, temp, ws);
    build_weff<<<(SUB_NO*2*T_NO + SUB_NO + 255)/256, 256, 0, stream>>>(conv_w, conv_b, ff_w, ff_b, ws);
    group_gemm_wmma<<<(T_DATA/16)/5, 160, 0, stream>>>(Se, Si, ws);
    conv_head<<<(T_DATA + TB - 1)/TB, TB, 0, stream>>>(ws, root_w, root_b, V_o, out);
}